// PointNet_54339926229280
// MI455X (gfx1250) — compile-verified
//
#include <hip/hip_runtime.h>
#include <hip/hip_bf16.h>

#define DEVFN __device__ __forceinline__

typedef __attribute__((ext_vector_type(2))) float v2f;
typedef __attribute__((ext_vector_type(8))) float v8f;

// ---------------- problem constants ----------------
constexpr int   NPTS  = 1000000;
constexpr int   SSEG  = 25000;
constexpr int   PTS   = 40;
constexpr int   ZSTR  = 12;                 // z row stride (8 feat + 3 pos_n + 1 ones)
constexpr int   NBLK1 = 489;                // blocks for N-sized passes (2048 rows each)
constexpr long  NPAD  = (long)NBLK1 * 2048; // 1,001,472
constexpr int   SP    = 25008;              // S padded to 16
constexpr int   GSTR  = 68;                 // g'/b1 row stride (64 feat + diam + one + 2 pad)
constexpr int   NSTRIP_S = SP / 16;         // 1563
constexpr int   GBLK  = (NSTRIP_S + 7) / 8; // 196
constexpr float EPS   = 1e-5f;
constexpr float SLOPE = 0.01f;

// ---------------- workspace layout (floats) ----------------
constexpr size_t OFF_Z    = 0;
constexpr size_t SZ_Z     = (size_t)NPAD * ZSTR;
constexpr size_t OFF_DIAM = OFF_Z + SZ_Z;
constexpr size_t OFF_W1P  = OFF_DIAM + 25024;
constexpr size_t OFF_W1F  = OFF_W1P + 12 * 64;
constexpr size_t OFF_W2F  = OFF_W1F + 12 * 64;
constexpr size_t OFF_C2   = OFF_W2F + 64 * 64;
constexpr size_t OFF_GW1P = OFF_C2 + 64;
constexpr size_t OFF_GW1F = OFF_GW1P + 68 * 64;
constexpr size_t OFF_GW2P = OFF_GW1F + 68 * 64;
constexpr size_t OFF_GW2F = OFF_GW2P + 68 * 32;
constexpr size_t OFF_P1   = OFF_GW2F + 68 * 32;
constexpr size_t OFF_P2   = OFF_P1 + (size_t)NBLK1 * 128;
constexpr size_t OFF_GP1  = OFF_P2 + (size_t)NBLK1 * 128;
constexpr size_t OFF_GP2  = OFF_GP1 + (size_t)GBLK * 128;
constexpr size_t OFF_G    = OFF_GP2 + (size_t)GBLK * 64;
constexpr size_t OFF_B1   = OFF_G + (size_t)SP * GSTR;
// total = OFF_B1 + SP*GSTR = ~15.63M floats (~62.5 MB)

// ---------------- helpers ----------------
DEVFN v8f wmma4(v2f a, v2f b, v8f c) {
  return __builtin_amdgcn_wmma_f32_16x16x4_f32(false, a, false, b, (short)0, c,
                                               false, false);
}
DEVFN float lrelu(float x) { return x > 0.f ? x : SLOPE * x; }
DEVFN void dswait() { asm volatile("s_wait_dscnt 0" ::: "memory"); }

// A fragment (16x4, f32): lane = M; VGPR0/1 = K,K+1 (lanes<16) or K+2,K+3
DEVFN v2f afrag(const float* base, int stride, int k0, int lane) {
  int m  = lane & 15;
  int kk = k0 + ((lane >> 4) << 1);
  const float* p = base + (long)m * stride + kk;
  v2f r; r.x = p[0]; r.y = p[1];
  return r;
}
// B fragment (4x16, f32) from row-major W[K][cols]: lane = N (mirrored)
DEVFN v2f bfrag(const float* W, int cols, int k0, int n0, int lane) {
  int n  = n0 + (lane & 15);
  int kk = k0 + ((lane >> 4) << 1);
  v2f r; r.x = W[kk * cols + n]; r.y = W[(kk + 1) * cols + n];
  return r;
}

// ============ K0: prep padded weights, zero pad rows ============
__global__ void k_prep(float* ws, const float* lw1, const float* gw1,
                       const float* gw2) {
  int i = blockIdx.x * 256 + threadIdx.x;
  constexpr int A = 768, B = A + 4352, C = B + 2176;
  constexpr int D = C + (int)(NPAD - NPTS) * ZSTR;       // z pad
  constexpr int E = D + (SP - SSEG) * GSTR;              // g' pad
  if (i < A) {
    int r = i >> 6; ws[OFF_W1P + i] = (r < 11) ? lw1[i] : 0.f;
  } else if (i < B) {
    int k = i - A; int r = k >> 6; ws[OFF_GW1P + k] = (r < 65) ? gw1[k] : 0.f;
  } else if (i < C) {
    int k = i - B; int r = k >> 5; ws[OFF_GW2P + k] = (r < 64) ? gw2[k] : 0.f;
  } else if (i < D) {
    int k = i - C; ws[OFF_Z + (size_t)NPTS * ZSTR + k] = 0.f;
  } else if (i < E) {
    int k = i - D; ws[OFF_G + (size_t)SSEG * GSTR + k] = 0.f;
  }
}

// ============ K1: per-segment unit-sphere normalize, build z ============
__global__ void k_segnorm(const float* pos, const float* x, float* ws) {
  __shared__ float red[64];
  __shared__ float cen[3];
  __shared__ float rad;
  float* z    = ws + OFF_Z;
  float* diam = ws + OFF_DIAM;
  int s = blockIdx.x, t = threadIdx.x;
  long p = (long)s * PTS + t;
  bool act = t < PTS;
  float px = 0.f, py = 0.f, pz = 0.f;
  if (act) { px = pos[p * 3]; py = pos[p * 3 + 1]; pz = pos[p * 3 + 2]; }
  for (int c = 0; c < 3; ++c) {
    red[t] = act ? (c == 0 ? px : (c == 1 ? py : pz)) : 0.f;
    __syncthreads();
    for (int off = 32; off > 0; off >>= 1) {
      if (t < off) red[t] += red[t + off];
      __syncthreads();
    }
    if (t == 0) cen[c] = red[0] / (float)PTS;
    __syncthreads();
  }
  float cx = px - cen[0], cy = py - cen[1], cz = pz - cen[2];
  red[t] = act ? (cx * cx + cy * cy + cz * cz) : 0.f;
  __syncthreads();
  for (int off = 32; off > 0; off >>= 1) {
    if (t < off) red[t] = fmaxf(red[t], red[t + off]);
    __syncthreads();
  }
  if (t == 0) {
    float r = sqrtf(fmaxf(red[0], 1e-12f));
    rad = r;
    diam[s] = 2.f * r;
  }
  __syncthreads();
  if (act) {
    float inv = 1.f / rad;
    float* zr = z + p * ZSTR;
    for (int j = 0; j < 8; ++j) zr[j] = x[p * 8 + j];
    zr[8] = cx * inv; zr[9] = cy * inv; zr[10] = cz * inv; zr[11] = 1.f;
  }
}

// ============ K2: stats of h1 = z @ W1p (raw) ============
__global__ void __launch_bounds__(256) k_stats1(const float* z,
                                                const float* W1p,
                                                float* part) {
  __shared__ float ldsS[8][128];
  int tid = threadIdx.x;
  for (int i = tid; i < 1024; i += 256) (&ldsS[0][0])[i] = 0.f;
  __syncthreads();
  int wave = tid >> 5, lane = tid & 31;
  v2f bf[3][4];  // [ks][nt]
  for (int ks = 0; ks < 3; ++ks)
    for (int nt = 0; nt < 4; ++nt)
      bf[ks][nt] = bfrag(W1p, 64, ks * 4, nt * 16, lane);
  float sA[4] = {0, 0, 0, 0}, qA[4] = {0, 0, 0, 0};
  long stripBase = ((long)blockIdx.x * 8 + wave) * 16;
#pragma unroll 1
  for (int i = 0; i < 16; ++i) {
    long row0 = (stripBase + i) * 16;
    const float* zb = z + row0 * ZSTR;
    v8f acc[4] = {{}, {}, {}, {}};
#pragma unroll
    for (int ks = 0; ks < 3; ++ks) {
      v2f a = afrag(zb, ZSTR, ks * 4, lane);
#pragma unroll
      for (int nt = 0; nt < 4; ++nt) acc[nt] = wmma4(a, bf[ks][nt], acc[nt]);
    }
#pragma unroll
    for (int nt = 0; nt < 4; ++nt)
      for (int r = 0; r < 8; ++r) { float e = acc[nt][r]; sA[nt] += e; qA[nt] += e * e; }
  }
  for (int nt = 0; nt < 4; ++nt) {
    float s = sA[nt] + __shfl_xor(sA[nt], 16);
    float q = qA[nt] + __shfl_xor(qA[nt], 16);
    if (lane < 16) { ldsS[wave][nt * 16 + lane] = s; ldsS[wave][64 + nt * 16 + lane] = q; }
  }
  __syncthreads();
  if (tid < 128) {
    float t = 0.f;
    for (int w = 0; w < 8; ++w) t += ldsS[w][tid];
    part[(size_t)blockIdx.x * 128 + tid] = t;
  }
}

// ============ K3: fold BN1 into W1f (ones-column carries bias) ============
__global__ void k_fold1(const float* part, const float* lw1, const float* g,
                        const float* b, float* W1f) {
  int j = threadIdx.x;  // 64
  float s = 0.f, q = 0.f;
  for (int bk = 0; bk < NBLK1; ++bk) { s += part[bk * 128 + j]; q += part[bk * 128 + 64 + j]; }
  float m  = s / (float)NPTS;
  float v  = q / (float)NPTS - m * m;
  float sc = g[j] * rsqrtf(v + EPS);
  for (int i = 0; i < 11; ++i) W1f[i * 64 + j] = lw1[i * 64 + j] * sc;
  W1f[11 * 64 + j] = b[j] - m * sc;
}

// ============ K4: a1 = act(z@W1f); stats of h2 = a1 @ lw2 ============
__global__ void __launch_bounds__(256) k_stats2(const float* z,
                                                const float* W1f,
                                                const float* lw2,
                                                float* part) {
  __shared__ float w2s[4096];
  __shared__ float tiles[8][16][68];
  __shared__ float ldsS[8][128];
  int tid = threadIdx.x;
  for (int i = tid; i < 4096; i += 256) w2s[i] = lw2[i];
  for (int i = tid; i < 1024; i += 256) (&ldsS[0][0])[i] = 0.f;
  __syncthreads();
  int wave = tid >> 5, lane = tid & 31;
  int m15 = lane & 15, hi8 = (lane >> 4) << 3;
  v2f bf[3][4];
  for (int ks = 0; ks < 3; ++ks)
    for (int nt = 0; nt < 4; ++nt)
      bf[ks][nt] = bfrag(W1f, 64, ks * 4, nt * 16, lane);
  float sA[4] = {0, 0, 0, 0}, qA[4] = {0, 0, 0, 0};
  long stripBase = ((long)blockIdx.x * 8 + wave) * 16;
  float* tb = &tiles[wave][0][0];
#pragma unroll 1
  for (int i = 0; i < 16; ++i) {
    long row0 = (stripBase + i) * 16;
    const float* zb = z + row0 * ZSTR;
    {
      v8f acc[4] = {{}, {}, {}, {}};
#pragma unroll
      for (int ks = 0; ks < 3; ++ks) {
        v2f a = afrag(zb, ZSTR, ks * 4, lane);
#pragma unroll
        for (int nt = 0; nt < 4; ++nt) acc[nt] = wmma4(a, bf[ks][nt], acc[nt]);
      }
#pragma unroll
      for (int nt = 0; nt < 4; ++nt)
        for (int r = 0; r < 8; ++r)
          tb[(r + hi8) * 68 + nt * 16 + m15] = lrelu(acc[nt][r]);
    }
    dswait();
    {
      v8f acc[4] = {{}, {}, {}, {}};
#pragma unroll 1
      for (int ks = 0; ks < 16; ++ks) {
        v2f a = afrag(tb, 68, ks * 4, lane);
#pragma unroll
        for (int nt = 0; nt < 4; ++nt)
          acc[nt] = wmma4(a, bfrag(w2s, 64, ks * 4, nt * 16, lane), acc[nt]);
      }
#pragma unroll
      for (int nt = 0; nt < 4; ++nt)
        for (int r = 0; r < 8; ++r) { float e = acc[nt][r]; sA[nt] += e; qA[nt] += e * e; }
    }
  }
  for (int nt = 0; nt < 4; ++nt) {
    float s = sA[nt] + __shfl_xor(sA[nt], 16);
    float q = qA[nt] + __shfl_xor(qA[nt], 16);
    if (lane < 16) { ldsS[wave][nt * 16 + lane] = s; ldsS[wave][64 + nt * 16 + lane] = q; }
  }
  __syncthreads();
  if (tid < 128) {
    float t = 0.f;
    for (int w = 0; w < 8; ++w) t += ldsS[w][tid];
    part[(size_t)blockIdx.x * 128 + tid] = t;
  }
}

// ============ K5: fold BN2 into W2f + bias c2 (epilogue add) ============
__global__ void k_fold2(const float* part, const float* lw2, const float* g,
                        const float* b, float* W2f, float* c2) {
  int j = threadIdx.x;  // 64
  float s = 0.f, q = 0.f;
  for (int bk = 0; bk < NBLK1; ++bk) { s += part[bk * 128 + j]; q += part[bk * 128 + 64 + j]; }
  float m  = s / (float)NPTS;
  float v  = q / (float)NPTS - m * m;
  float sc = g[j] * rsqrtf(v + EPS);
  for (int i = 0; i < 64; ++i) W2f[i * 64 + j] = lw2[i * 64 + j] * sc;
  c2[j] = b[j] - m * sc;
}

// ============ K6: recompute a1,a2; per-segment max pool -> g' ============
__global__ void __launch_bounds__(256) k_pool(const float* z, const float* W1f,
                                              const float* W2f, const float* c2,
                                              const float* diam, float* gp) {
  __shared__ float w2s[4096];
  __shared__ float tiles[8][16][68];
  int tid = threadIdx.x;
  for (int i = tid; i < 4096; i += 256) w2s[i] = W2f[i];
  __syncthreads();
  int wave = tid >> 5, lane = tid & 31;
  int m15 = lane & 15, hi8 = (lane >> 4) << 3;
  long chunk = (long)blockIdx.x * 8 + wave;  // 2 segments / chunk
  if (chunk < SSEG / 2) {
    v2f bf[3][4];
    for (int ks = 0; ks < 3; ++ks)
      for (int nt = 0; nt < 4; ++nt)
        bf[ks][nt] = bfrag(W1f, 64, ks * 4, nt * 16, lane);
    float c2v[4];
    for (int nt = 0; nt < 4; ++nt) c2v[nt] = c2[nt * 16 + m15];
    float pm[4][2];
    for (int nt = 0; nt < 4; ++nt) { pm[nt][0] = -3.4e38f; pm[nt][1] = -3.4e38f; }
    float* tb = &tiles[wave][0][0];
#pragma unroll 1
    for (int st = 0; st < 5; ++st) {
      long row0 = chunk * 80 + (long)st * 16;
      const float* zb = z + row0 * ZSTR;
      {
        v8f acc[4] = {{}, {}, {}, {}};
#pragma unroll
        for (int ks = 0; ks < 3; ++ks) {
          v2f a = afrag(zb, ZSTR, ks * 4, lane);
#pragma unroll
          for (int nt = 0; nt < 4; ++nt) acc[nt] = wmma4(a, bf[ks][nt], acc[nt]);
        }
#pragma unroll
        for (int nt = 0; nt < 4; ++nt)
          for (int r = 0; r < 8; ++r)
            tb[(r + hi8) * 68 + nt * 16 + m15] = lrelu(acc[nt][r]);
      }
      dswait();
      {
        v8f acc[4] = {{}, {}, {}, {}};
#pragma unroll 1
        for (int ks = 0; ks < 16; ++ks) {
          v2f a = afrag(tb, 68, ks * 4, lane);
#pragma unroll
          for (int nt = 0; nt < 4; ++nt)
            acc[nt] = wmma4(a, bfrag(w2s, 64, ks * 4, nt * 16, lane), acc[nt]);
        }
#pragma unroll
        for (int nt = 0; nt < 4; ++nt)
          for (int r = 0; r < 8; ++r) {
            float e = lrelu(acc[nt][r] + c2v[nt]);
            int ric = st * 16 + r + hi8;
            int sg = (ric >= PTS) ? 1 : 0;
            pm[nt][sg] = fmaxf(pm[nt][sg], e);
          }
      }
    }
    for (int nt = 0; nt < 4; ++nt)
      for (int sg = 0; sg < 2; ++sg)
        pm[nt][sg] = fmaxf(pm[nt][sg], __shfl_xor(pm[nt][sg], 16));
    if (lane < 16) {
      for (int sg = 0; sg < 2; ++sg) {
        long srow = chunk * 2 + sg;
        for (int nt = 0; nt < 4; ++nt)
          gp[srow * GSTR + nt * 16 + lane] = pm[nt][sg];
      }
    }
    if (lane == 0) {
      for (int sg = 0; sg < 2; ++sg) {
        long srow = chunk * 2 + sg;
        gp[srow * GSTR + 64] = diam[srow];
        gp[srow * GSTR + 65] = 1.f;
        gp[srow * GSTR + 66] = 0.f;
        gp[srow * GSTR + 67] = 0.f;
      }
    }
  }
}

// ============ K7: stats of hg1 = g' @ gw1p (raw) ============
__global__ void __launch_bounds__(256) k_gstats1(const float* gp,
                                                 const float* GW1p,
                                                 float* part) {
  __shared__ float w1s[4352];
  __shared__ float ldsS[8][128];
  int tid = threadIdx.x;
  for (int i = tid; i < 4352; i += 256) w1s[i] = GW1p[i];
  for (int i = tid; i < 1024; i += 256) (&ldsS[0][0])[i] = 0.f;
  __syncthreads();
  int wave = tid >> 5, lane = tid & 31;
  int strip = blockIdx.x * 8 + wave;
  if (strip < NSTRIP_S) {
    long row0 = (long)strip * 16;
    const float* ab = gp + row0 * GSTR;
    v8f acc[4] = {{}, {}, {}, {}};
#pragma unroll 1
    for (int ks = 0; ks < 17; ++ks) {
      v2f a = afrag(ab, GSTR, ks * 4, lane);
#pragma unroll
      for (int nt = 0; nt < 4; ++nt)
        acc[nt] = wmma4(a, bfrag(w1s, 64, ks * 4, nt * 16, lane), acc[nt]);
    }
    for (int nt = 0; nt < 4; ++nt) {
      float sA = 0.f, qA = 0.f;
      for (int r = 0; r < 8; ++r) { float e = acc[nt][r]; sA += e; qA += e * e; }
      float s = sA + __shfl_xor(sA, 16);
      float q = qA + __shfl_xor(qA, 16);
      if (lane < 16) { ldsS[wave][nt * 16 + lane] = s; ldsS[wave][64 + nt * 16 + lane] = q; }
    }
  }
  __syncthreads();
  if (tid < 128) {
    float t = 0.f;
    for (int w = 0; w < 8; ++w) t += ldsS[w][tid];
    part[(size_t)blockIdx.x * 128 + tid] = t;
  }
}

// ============ K8: fold gBN1 into gw1f ============
__global__ void k_gfold1(const float* part, const float* gw1, const float* g,
                         const float* b, float* W1f) {
  int j = threadIdx.x;  // 64
  float s = 0.f, q = 0.f;
  for (int bk = 0; bk < GBLK; ++bk) { s += part[bk * 128 + j]; q += part[bk * 128 + 64 + j]; }
  float m  = s / (float)SSEG;
  float v  = q / (float)SSEG - m * m;
  float sc = g[j] * rsqrtf(v + EPS);
  for (int i = 0; i < 65; ++i) W1f[i * 64 + j] = gw1[i * 64 + j] * sc;
  W1f[65 * 64 + j] = b[j] - m * sc;
  W1f[66 * 64 + j] = 0.f;
  W1f[67 * 64 + j] = 0.f;
}

// ============ K9: b1 = act(g'@gw1f); stats of hg2 = b1 @ gw2p ============
__global__ void __launch_bounds__(256) k_gstats2(const float* gp,
                                                 const float* W1fg,
                                                 const float* W2pg, float* b1,
                                                 float* part) {
  __shared__ float w1s[4352];
  __shared__ float w2s[2176];
  __shared__ float tiles[8][16][68];
  __shared__ float ldsS[8][64];
  int tid = threadIdx.x;
  for (int i = tid; i < 4352; i += 256) w1s[i] = W1fg[i];
  for (int i = tid; i < 2176; i += 256) w2s[i] = W2pg[i];
  for (int i = tid; i < 512; i += 256) (&ldsS[0][0])[i] = 0.f;
  __syncthreads();
  int wave = tid >> 5, lane = tid & 31;
  int m15 = lane & 15, hi8 = (lane >> 4) << 3;
  int strip = blockIdx.x * 8 + wave;
  if (strip < NSTRIP_S) {
    long row0 = (long)strip * 16;
    const float* ab = gp + row0 * GSTR;
    float* tb = &tiles[wave][0][0];
    {
      v8f acc[4] = {{}, {}, {}, {}};
#pragma unroll 1
      for (int ks = 0; ks < 17; ++ks) {
        v2f a = afrag(ab, GSTR, ks * 4, lane);
#pragma unroll
        for (int nt = 0; nt < 4; ++nt)
          acc[nt] = wmma4(a, bfrag(w1s, 64, ks * 4, nt * 16, lane), acc[nt]);
      }
#pragma unroll
      for (int nt = 0; nt < 4; ++nt)
        for (int r = 0; r < 8; ++r) {
          float e = lrelu(acc[nt][r]);
          int M = r + hi8;
          tb[M * 68 + nt * 16 + m15] = e;
          b1[(row0 + M) * GSTR + nt * 16 + m15] = e;
        }
    }
    if (lane < 16) {
      int M = lane;
      long row = row0 + M;
      float one = (row < SSEG) ? 1.f : 0.f;
      tb[M * 68 + 64] = one; tb[M * 68 + 65] = 0.f;
      tb[M * 68 + 66] = 0.f; tb[M * 68 + 67] = 0.f;
      b1[row * GSTR + 64] = one; b1[row * GSTR + 65] = 0.f;
      b1[row * GSTR + 66] = 0.f; b1[row * GSTR + 67] = 0.f;
    }
    dswait();
    {
      v8f acc[2] = {{}, {}};
#pragma unroll 1
      for (int ks = 0; ks < 17; ++ks) {
        v2f a = afrag(tb, 68, ks * 4, lane);
#pragma unroll
        for (int nt = 0; nt < 2; ++nt)
          acc[nt] = wmma4(a, bfrag(w2s, 32, ks * 4, nt * 16, lane), acc[nt]);
      }
      for (int nt = 0; nt < 2; ++nt) {
        float sA = 0.f, qA = 0.f;
        for (int r = 0; r < 8; ++r) { float e = acc[nt][r]; sA += e; qA += e * e; }
        float s = sA + __shfl_xor(sA, 16);
        float q = qA + __shfl_xor(qA, 16);
        if (lane < 16) { ldsS[wave][nt * 16 + lane] = s; ldsS[wave][32 + nt * 16 + lane] = q; }
      }
    }
  }
  __syncthreads();
  if (tid < 64) {
    float t = 0.f;
    for (int w = 0; w < 8; ++w) t += ldsS[w][tid];
    part[(size_t)blockIdx.x * 64 + tid] = t;
  }
}

// ============ K10: fold gBN2 into gw2f ============
__global__ void k_gfold2(const float* part, const float* gw2, const float* g,
                         const float* b, float* W2f) {
  int j = threadIdx.x;  // 32
  float s = 0.f, q = 0.f;
  for (int bk = 0; bk < GBLK; ++bk) { s += part[bk * 64 + j]; q += part[bk * 64 + 32 + j]; }
  float m  = s / (float)SSEG;
  float v  = q / (float)SSEG - m * m;
  float sc = g[j] * rsqrtf(v + EPS);
  for (int i = 0; i < 64; ++i) W2f[i * 32 + j] = gw2[i * 32 + j] * sc;
  W2f[64 * 32 + j] = b[j] - m * sc;
  W2f[65 * 32 + j] = 0.f;
  W2f[66 * 32 + j] = 0.f;
  W2f[67 * 32 + j] = 0.f;
}

// ============ K11: b2 = act(b1@gw2f); out = b2 @ hw + hb ============
__global__ void __launch_bounds__(256) k_head(const float* b1,
                                              const float* W2fg,
                                              const float* hw, const float* hb,
                                              float* out) {
  __shared__ float w2s[2176];
  __shared__ float bt[8][16][36];
  int tid = threadIdx.x;
  for (int i = tid; i < 2176; i += 256) w2s[i] = W2fg[i];
  __syncthreads();
  int wave = tid >> 5, lane = tid & 31;
  int m15 = lane & 15, hi8 = (lane >> 4) << 3;
  int strip = blockIdx.x * 8 + wave;
  if (strip < NSTRIP_S) {
    long row0 = (long)strip * 16;
    const float* ab = b1 + row0 * GSTR;
    float* tb = &bt[wave][0][0];
    {
      v8f acc[2] = {{}, {}};
#pragma unroll 1
      for (int ks = 0; ks < 17; ++ks) {
        v2f a = afrag(ab, GSTR, ks * 4, lane);
#pragma unroll
        for (int nt = 0; nt < 2; ++nt)
          acc[nt] = wmma4(a, bfrag(w2s, 32, ks * 4, nt * 16, lane), acc[nt]);
      }
#pragma unroll
      for (int nt = 0; nt < 2; ++nt)
        for (int r = 0; r < 8; ++r)
          tb[(r + hi8) * 36 + nt * 16 + m15] = lrelu(acc[nt][r]);
    }
    dswait();
    for (int t = 0; t < 7; ++t) {
      int k = t * 32 + lane;
      if (k < 208) {
        int r = k / 13, c = k % 13;
        long row = row0 + r;
        if (row < SSEG) {
          float acc = hb[c];
          for (int j = 0; j < 32; ++j) acc += tb[r * 36 + j] * hw[j * 13 + c];
          out[row * 13 + c] = acc;
        }
      }
    }
  }
}

// ================= launcher =================
extern "C" void kernel_launch(void* const* d_in, const int* in_sizes, int n_in,
                              void* d_out, int out_size, void* d_ws,
                              size_t ws_size, hipStream_t stream) {
  const float* pos = (const float*)d_in[0];
  const float* x   = (const float*)d_in[1];
  const float* lw1 = (const float*)d_in[2];
  const float* lg1 = (const float*)d_in[3];
  const float* lb1 = (const float*)d_in[4];
  const float* lw2 = (const float*)d_in[5];
  const float* lg2 = (const float*)d_in[6];
  const float* lb2 = (const float*)d_in[7];
  const float* gw1 = (const float*)d_in[8];
  const float* gg1 = (const float*)d_in[9];
  const float* gb1 = (const float*)d_in[10];
  const float* gw2 = (const float*)d_in[11];
  const float* gg2 = (const float*)d_in[12];
  const float* gb2 = (const float*)d_in[13];
  const float* hw  = (const float*)d_in[14];
  const float* hb  = (const float*)d_in[15];
  float* ws = (float*)d_ws;
  float* outp = (float*)d_out;

  k_prep<<<100, 256, 0, stream>>>(ws, lw1, gw1, gw2);
  k_segnorm<<<SSEG, 64, 0, stream>>>(pos, x, ws);
  k_stats1<<<NBLK1, 256, 0, stream>>>(ws + OFF_Z, ws + OFF_W1P, ws + OFF_P1);
  k_fold1<<<1, 64, 0, stream>>>(ws + OFF_P1, lw1, lg1, lb1, ws + OFF_W1F);
  k_stats2<<<NBLK1, 256, 0, stream>>>(ws + OFF_Z, ws + OFF_W1F, lw2, ws + OFF_P2);
  k_fold2<<<1, 64, 0, stream>>>(ws + OFF_P2, lw2, lg2, lb2, ws + OFF_W2F, ws + OFF_C2);
  k_pool<<<(SSEG / 2 + 7) / 8, 256, 0, stream>>>(ws + OFF_Z, ws + OFF_W1F,
                                                 ws + OFF_W2F, ws + OFF_C2,
                                                 ws + OFF_DIAM, ws + OFF_G);
  k_gstats1<<<GBLK, 256, 0, stream>>>(ws + OFF_G, ws + OFF_GW1P, ws + OFF_GP1);
  k_gfold1<<<1, 64, 0, stream>>>(ws + OFF_GP1, gw1, gg1, gb1, ws + OFF_GW1F);
  k_gstats2<<<GBLK, 256, 0, stream>>>(ws + OFF_G, ws + OFF_GW1F, ws + OFF_GW2P,
                                      ws + OFF_B1, ws + OFF_GP2);
  k_gfold2<<<1, 32, 0, stream>>>(ws + OFF_GP2, gw2, gg2, gb2, ws + OFF_GW2F);
  k_head<<<GBLK, 256, 0, stream>>>(ws + OFF_B1, ws + OFF_GW2F, hw, hb, outp);
}